// Loss_79688823210509
// MI455X (gfx1250) — compile-verified
//
#include <hip/hip_runtime.h>
#include <stdint.h>

// ---------------------------------------------------------------------------
// YOLO-style 3-level loss for MI455X (gfx1250, wave32).
//   - memory-bound streaming reduction (~52MB -> ~2.3us at 23.3TB/s)
//   - TDM (tensor_load_to_lds) stages per-batch gt boxes into LDS
//   - division-free iou_max<0.5 test in the 150-box hot loop
//   - deterministic two-phase reduction; final sum via V_WMMA_F32_16X16X4_F32
// ---------------------------------------------------------------------------

typedef float v2f  __attribute__((ext_vector_type(2)));
typedef float v8f  __attribute__((ext_vector_type(8)));
typedef unsigned int u32x4 __attribute__((ext_vector_type(4)));
typedef int   i32x4 __attribute__((ext_vector_type(4)));
typedef int   i32x8 __attribute__((ext_vector_type(8)));

#define NBOX  150
#define NCLS  20
#define NBATCH 8
#define BLK   256
#define NWAVE (BLK / 32)

#if defined(__HIP_DEVICE_COMPILE__)
#  if __has_builtin(__builtin_amdgcn_tensor_load_to_lds) && \
      __has_builtin(__builtin_amdgcn_s_wait_tensorcnt)
#    define HAVE_TDM 1
#  endif
#  if __has_builtin(__builtin_amdgcn_wmma_f32_16x16x4_f32)
#    define HAVE_WMMA4 1
#  endif
#endif

__device__ __forceinline__ float sl1(float x, float t) {
  // smooth_l1 with BETA = 1/9: n<BETA ? 0.5*n*n/BETA : n - 0.5*BETA
  float n = fabsf(x - t);
  return n < (1.0f / 9.0f) ? 4.5f * n * n : n - (1.0f / 18.0f);
}
__device__ __forceinline__ float softplus_f(float x) {
  // stable softplus: max(x,0) + log1p(exp(-|x|))  (matches jax.nn.softplus)
  return fmaxf(x, 0.0f) + log1pf(expf(-fabsf(x)));
}

// ---------------------------------------------------------------------------
// One pyramid level. grid = (ceil(GG3/BLK), B); one anchor per thread.
//   p   : [B, GG3, 30]  (only channels 9..29 used)
//   pd  : [B, GG3, 9]
//   lab : [B, GG3, 33]
//   bb  : [B, 150, 4]   (xywh)
// Per-block partial row {s_iou, s_conf, s_cls, s_other}*(1/B) written to
// `partial` (deterministic path) or atomically added to `aout` (fallback).
// ---------------------------------------------------------------------------
__global__ __launch_bounds__(BLK) void yolo_layer_loss(
    const float* __restrict__ p, const float* __restrict__ pd,
    const float* __restrict__ lab, const float* __restrict__ bb,
    int GG3, float* __restrict__ partial, float* __restrict__ aout)
{
  __shared__ __align__(16) float raw[NBOX * 4];
  __shared__ float bx1[NBOX], by1[NBOX], bx2[NBOX], by2[NBOX], bar[NBOX];
  __shared__ float red[NWAVE][4];

  const int b   = blockIdx.y;
  const int tid = threadIdx.x;

  // ---- stage this batch's boxes into LDS (TDM async DMA if available) ----
#if defined(HAVE_TDM)
  if (tid < 32) {  // one wave issues the DMA (TDM ignores EXEC, issue once)
    uint64_t ga = (uint64_t)(uintptr_t)(bb + (size_t)b * (NBOX * 4));
    uint32_t lds_off = (uint32_t)(uintptr_t)(&raw[0]);  // low 32b = LDS offset
    u32x4 g0; i32x8 g1;
    i32x4 g2 = {0, 0, 0, 0}, g3 = {0, 0, 0, 0};
    i32x8 g4 = {0, 0, 0, 0, 0, 0, 0, 0};
    // D# group0: count=1 (valid), lds_addr, global_addr[56:0], type=2
    g0[0] = 1u;
    g0[1] = lds_off;
    g0[2] = (uint32_t)ga;
    g0[3] = (uint32_t)((ga >> 32) & 0x1FFFFFFu) | (2u << 30);
    // D# group1: data_size=4B; 1-row tile of 600 elements
    g1[0] = (2 << 16);                 // workgroup_mask=0, data_size=2 (4B)
    g1[1] = (int)((NBOX * 4) << 16);   // tensor_dim0 = 600 (bits 63:48)
    g1[2] = (1 << 16);                 // tensor_dim1 = 1   (bits 95:80)
    g1[3] = (int)((NBOX * 4) << 16);   // tile_dim0   = 600 (bits 127:112)
    g1[4] = 1;                         // tile_dim1 = 1, tile_dim2 = 0
    g1[5] = NBOX * 4;                  // tensor_dim0_stride = 600
    g1[6] = 0; g1[7] = 0;
    // amdgpu-toolchain (clang-23) 6-arg form: extra zero i32x8 before cpol
    __builtin_amdgcn_tensor_load_to_lds(g0, g1, g2, g3, g4, 0);
    __builtin_amdgcn_s_wait_tensorcnt(0);
  }
#else
  for (int j = tid; j < NBOX * 4; j += BLK)
    raw[j] = bb[(size_t)b * (NBOX * 4) + j];
#endif
  __syncthreads();

  // convert once to xyxy+area SoA (inner loop does all-lane LDS broadcasts)
  for (int j = tid; j < NBOX; j += BLK) {
    float cx = raw[4 * j], cy = raw[4 * j + 1], w = raw[4 * j + 2], h = raw[4 * j + 3];
    float x1 = cx - 0.5f * w, y1 = cy - 0.5f * h;
    float x2 = cx + 0.5f * w, y2 = cy + 0.5f * h;
    bx1[j] = x1; by1[j] = y1; bx2[j] = x2; by2[j] = y2;
    bar[j] = (x2 - x1) * (y2 - y1);
  }
  __syncthreads();

  float acc_iou = 0.f, acc_conf = 0.f, acc_cls = 0.f, acc_oth = 0.f;

  const int idx = blockIdx.x * BLK + tid;
  if (idx < GG3) {
    const size_t gi = (size_t)b * (size_t)GG3 + (size_t)idx;
    const float* PD = pd  + gi * 9;
    const float* L  = lab + gi * 33;
    const float* P  = p   + gi * 30;

    float px = PD[0], py = PD[1], pw = PD[2], ph = PD[3];
    float pa0 = PD[4], pa1 = PD[5], pa2 = PD[6], pa3 = PD[7], pr = PD[8];
    float lx = L[0], ly = L[1], lw = L[2], lh = L[3];
    float la0 = L[4], la1 = L[5], la2 = L[6], la3 = L[7], lr8 = L[8];
    float ls13 = L[9], ls24 = L[10], obj = L[11], mix = L[12];
    float pconf = P[9];

    // xyxy + areas
    float ax1 = px - 0.5f * pw, ay1 = py - 0.5f * ph;
    float ax2 = px + 0.5f * pw, ay2 = py + 0.5f * ph;
    float gx1 = lx - 0.5f * lw, gy1 = ly - 0.5f * lh;
    float gx2 = lx + 0.5f * lw, gy2 = ly + 0.5f * lh;
    float a1 = (ax2 - ax1) * (ay2 - ay1);
    float a2 = (gx2 - gx1) * (gy2 - gy1);

    // GIoU(pred, label)
    float tlx = fmaxf(ax1, gx1), tly = fmaxf(ay1, gy1);
    float brx = fminf(ax2, gx2), bry = fminf(ay2, gy2);
    float inter = fmaxf(brx - tlx, 0.f) * fmaxf(bry - tly, 0.f);
    float uni = a1 + a2 - inter;
    float iou = inter / (uni + 1e-9f);
    float cw = fmaxf(ax2, gx2) - fminf(ax1, gx1);
    float chh = fmaxf(ay2, gy2) - fminf(ay1, gy1);
    float carea = cw * chh;
    float giou = iou - (carea - uni) / (carea + 1e-9f);

    const float img2inv = 1.0f / (608.0f * 608.0f);  // (stride*G)^2 identical all levels
    float whf = lw * lh * img2inv;
    float om = obj * mix;

    float l_iou = om * (2.0f - whf) * (1.0f - giou);

    float suma = sl1(pa0, la0) + sl1(pa1, la1) + sl1(pa2, la2) + sl1(pa3, la3);
    float l_a = om * (1.0f + whf) * suma;
    float l_r = om * 16.0f * sl1(pr, lr8);

    float d13 = pw * fabsf(1.0f - pa0 - pa2);
    float s13 = ph / sqrtf(d13 * d13 + ph * ph);
    float d24 = ph * fabsf(1.0f - pa1 - pa3);
    float s24 = pw / sqrtf(d24 * d24 + pw * pw);
    float l_s = om * (sl1(s13, ls13) / (ls13 + 1e-8f) +
                      sl1(s24, ls24) / (ls24 + 1e-8f));

    // focal conf
    float sig = 1.0f / (1.0f + expf(-pconf));
    float dlt = obj - sig;
    float focal = (softplus_f(pconf) - pconf * obj) * dlt * dlt;

    // noobj: max_j iou_j < 0.5  <=>  max_j (3*inter_j - area_j) < a1 + 1e-9
    // (division-free; union>0 since all boxes have positive area)
    float m = -3.0e38f;
#pragma unroll 5
    for (int j = 0; j < NBOX; ++j) {
      float ix = fminf(ax2, bx2[j]) - fmaxf(ax1, bx1[j]);
      float iy = fminf(ay2, by2[j]) - fmaxf(ay1, by1[j]);
      float it = fmaxf(ix, 0.f) * fmaxf(iy, 0.f);
      m = fmaxf(m, fmaf(3.0f, it, -bar[j]));
    }
    float noobj = (1.0f - obj) * ((m < a1 + 1e-9f) ? 1.0f : 0.0f);
    float l_conf = (obj + noobj) * focal * mix;

    float scls = 0.f;
#pragma unroll
    for (int c = 0; c < NCLS; ++c) {
      float x = P[10 + c], t = L[13 + c];
      scls += softplus_f(x) - x * t;
    }
    float l_cls = om * scls;

    acc_iou = l_iou;
    acc_conf = l_conf;
    acc_cls = l_cls;
    acc_oth = l_a + l_r + l_s;
  }

  // ---- wave32 reduce, then cross-wave via LDS ----
#pragma unroll
  for (int o = 16; o > 0; o >>= 1) {
    acc_iou  += __shfl_down(acc_iou,  o, 32);
    acc_conf += __shfl_down(acc_conf, o, 32);
    acc_cls  += __shfl_down(acc_cls,  o, 32);
    acc_oth  += __shfl_down(acc_oth,  o, 32);
  }
  const int wave = tid >> 5, lane = tid & 31;
  if (lane == 0) {
    red[wave][0] = acc_iou; red[wave][1] = acc_conf;
    red[wave][2] = acc_cls; red[wave][3] = acc_oth;
  }
  __syncthreads();
  if (tid == 0) {
    float si = 0.f, sc = 0.f, sl = 0.f, so = 0.f;
#pragma unroll
    for (int w = 0; w < NWAVE; ++w) {
      si += red[w][0]; sc += red[w][1]; sl += red[w][2]; so += red[w][3];
    }
    si *= 0.125f; sc *= 0.125f; sl *= 0.125f; so *= 0.125f;  // 1/B
    if (partial) {
      float* row = partial + (size_t)(blockIdx.y * gridDim.x + blockIdx.x) * 4;
      row[0] = si; row[1] = sc; row[2] = sl; row[3] = so;
    } else {
      atomicAdd(&aout[0], si + sc + sl + so);
      atomicAdd(&aout[1], si);
      atomicAdd(&aout[2], sc);
      atomicAdd(&aout[3], sl);
    }
  }
}

// ---------------------------------------------------------------------------
// Finalize: sum P partial rows (P%4==0, pad rows zeroed) -> 4 outputs.
// ones(16x4) x B gives D[m][n] = sum_k B[k][n]; each WMMA folds 4 rows into
// columns 0..3 and accumulates through C. Single wave, deterministic.
// ---------------------------------------------------------------------------
__global__ __launch_bounds__(32) void finalize_k(
    const float* __restrict__ partial, int nrows, float* __restrict__ out)
{
#if defined(HAVE_WMMA4)
  const int lane = threadIdx.x;
  const int n = lane & 15;
  const int hi = lane >> 4;  // lane half selects 2 of the 4 chunk rows
  v2f a; a[0] = 1.0f; a[1] = 1.0f;  // A = ones(16x4)
  v8f c = {};
  for (int r0 = 0; r0 < nrows; r0 += 4) {
    v2f bm; bm[0] = 0.f; bm[1] = 0.f;
    if (n < 4) {
      const float* base = partial + (size_t)(r0 + 2 * hi) * 4 + n;
      bm[0] = base[0];  // row r0 + 2*hi
      bm[1] = base[4];  // row r0 + 2*hi + 1
    }
    c = __builtin_amdgcn_wmma_f32_16x16x4_f32(
        false, a, false, bm, (short)0, c, false, false);
  }
  // c[0] on lanes 0..15 holds D[0][lane]; columns 0..3 are the sums
  __shared__ float s[4];
  if (lane < 4) s[lane] = c[0];
  __syncthreads();
  if (lane == 0) {
    out[0] = s[0] + s[1] + s[2] + s[3];  // total
    out[1] = s[0];                       // s_iou
    out[2] = s[1];                       // s_conf
    out[3] = s[2];                       // s_cls
  }
#else
  const int lane = threadIdx.x;
  float s0 = 0.f, s1 = 0.f, s2 = 0.f, s3 = 0.f;
  for (int r = lane; r < nrows; r += 32) {
    s0 += partial[r * 4 + 0]; s1 += partial[r * 4 + 1];
    s2 += partial[r * 4 + 2]; s3 += partial[r * 4 + 3];
  }
#pragma unroll
  for (int o = 16; o > 0; o >>= 1) {
    s0 += __shfl_down(s0, o, 32); s1 += __shfl_down(s1, o, 32);
    s2 += __shfl_down(s2, o, 32); s3 += __shfl_down(s3, o, 32);
  }
  if (lane == 0) {
    out[0] = s0 + s1 + s2 + s3;
    out[1] = s0; out[2] = s1; out[3] = s2;
  }
#endif
}

__global__ void init_k(float* partial, int nfloats, float* out)
{
  int i = blockIdx.x * blockDim.x + threadIdx.x;
  if (i < 4) out[i] = 0.f;
  if (partial)
    for (int j = i; j < nfloats; j += blockDim.x * gridDim.x) partial[j] = 0.f;
}

extern "C" void kernel_launch(void* const* d_in, const int* in_sizes, int n_in,
                              void* d_out, int out_size, void* d_ws, size_t ws_size,
                              hipStream_t stream) {
  (void)in_sizes; (void)n_in; (void)out_size;
  // setup_inputs() dict order: p_s, pd_s, label_s, sbboxes, p_m, pd_m,
  // label_m, mbboxes, p_l, pd_l, label_l, lbboxes
  const float* p_s   = (const float*)d_in[0];
  const float* pd_s  = (const float*)d_in[1];
  const float* lab_s = (const float*)d_in[2];
  const float* bb_s  = (const float*)d_in[3];
  const float* p_m   = (const float*)d_in[4];
  const float* pd_m  = (const float*)d_in[5];
  const float* lab_m = (const float*)d_in[6];
  const float* bb_m  = (const float*)d_in[7];
  const float* p_l   = (const float*)d_in[8];
  const float* pd_l  = (const float*)d_in[9];
  const float* lab_l = (const float*)d_in[10];
  const float* bb_l  = (const float*)d_in[11];
  float* out = (float*)d_out;

  const int GG3_s = 76 * 76 * 3, GG3_m = 38 * 38 * 3, GG3_l = 19 * 19 * 3;
  const int nb_s = (GG3_s + BLK - 1) / BLK;  // 68
  const int nb_m = (GG3_m + BLK - 1) / BLK;  // 17
  const int nb_l = (GG3_l + BLK - 1) / BLK;  // 5
  const int rows = (nb_s + nb_m + nb_l) * NBATCH;  // 720
  const int rows_pad = (rows + 3) & ~3;

  const bool use_ws = ws_size >= (size_t)rows_pad * 4 * sizeof(float);
  float* partial = use_ws ? (float*)d_ws : nullptr;

  init_k<<<1, 256, 0, stream>>>(partial, rows_pad * 4, out);

  dim3 gs(nb_s, NBATCH), gm(nb_m, NBATCH), gl(nb_l, NBATCH);
  float* ps = partial;
  float* pm = partial ? partial + (size_t)nb_s * NBATCH * 4 : nullptr;
  float* pl = partial ? partial + (size_t)(nb_s + nb_m) * NBATCH * 4 : nullptr;
  yolo_layer_loss<<<gs, BLK, 0, stream>>>(p_s, pd_s, lab_s, bb_s, GG3_s, ps, out);
  yolo_layer_loss<<<gm, BLK, 0, stream>>>(p_m, pd_m, lab_m, bb_m, GG3_m, pm, out);
  yolo_layer_loss<<<gl, BLK, 0, stream>>>(p_l, pd_l, lab_l, bb_l, GG3_l, pl, out);

  if (use_ws) finalize_k<<<1, 32, 0, stream>>>(partial, rows_pad, out);
}